// CondLaneHead_26508538151626
// MI455X (gfx1250) — compile-verified
//
#include <hip/hip_runtime.h>

typedef __attribute__((ext_vector_type(2))) float v2f;
typedef __attribute__((ext_vector_type(8))) float v8f;

#define CH        64
#define CIN1      66
#define WW        200
#define HWPX      16000            // 80*200
#define NPARAM    8513
#define W1_OFF    0
#define W2_OFF    4224
#define W3_OFF    8320
#define B1_OFF    8384
#define B2_OFF    8448
#define B3_OFF    8512

#define WAVES     4
#define THREADS   (WAVES*32)
#define TPW       8                // pixel tiles per wave
#define TILES_WG  (WAVES*TPW)      // 32 tiles (512 px) per WG
#define WGS_X     32               // 32*32*16 = 16384 >= 16000 px
#define STAGE_W   1088             // 34 row-pairs * 32 words per wave

// ---------------------------------------------------------------------------
// Per-instance dynamic 3-layer MLP over pixels, built on V_WMMA_F32_16X16X4_F32.
// Assumed (ISA 7.12.2) f32 layouts, wave32:
//   A 16x4 : v0 = A[M=lane%16][K=2*hi],   v1 = A[..][K=2*hi+1]
//   B 4x16 : v0 = B[K=2*hi][N=lane%16],   v1 = B[K=2*hi+1][..]
//   C/D    : vgpr r, lane -> D[M = r + 8*hi (per 16-tile)][N = lane%16]
// Activations staged in LDS as row-pair interleaved: addr(row,px) =
//   (row>>1)*32 + 2*px + (row&1)  -> every B fragment is one aligned b64 load.
// ---------------------------------------------------------------------------
__global__ __launch_bounds__(THREADS)
void condlane_mlp_wmma(const float* __restrict__ x,
                       const float* __restrict__ params,
                       const int*   __restrict__ num_ins,
                       float*       __restrict__ out,
                       int n_images)
{
    __shared__ float ldsP[NPARAM + 3];          // per-instance params
    __shared__ float ldsS[WAVES * STAGE_W];     // per-wave activation stage

    const int inst = blockIdx.y;
    // instance -> image (jnp.repeat semantics over num_ins)
    int img = 0;
    {
        int acc = 0;
        for (int i = 0; i < n_images; ++i) {
            acc += num_ins[i];
            if (inst < acc) { img = i; break; }
        }
    }

    // cooperative param load (one-time per WG; L2-resident source)
    const float* pg = params + (size_t)inst * NPARAM;
    for (int i = threadIdx.x; i < NPARAM; i += THREADS) ldsP[i] = pg[i];

    const int lane = threadIdx.x & 31;
    const int wave = threadIdx.x >> 5;
    const int pxl  = lane & 15;                 // pixel within tile / N index
    const int hi   = lane >> 4;                 // wave half
    float* stage = &ldsS[wave * STAGE_W];
    stage[33 * 32 + lane] = 0.0f;               // zero K-pad rows 66,67 (pair 33)

    __syncthreads();

    const float* xb = x + (size_t)img * CH * HWPX;

    // preload per-lane W3 values (layer-3 dot done in VALU; M=1 wastes WMMA)
    float w3v[4][8];
    #pragma unroll
    for (int m = 0; m < 4; ++m)
        #pragma unroll
        for (int r = 0; r < 8; ++r)
            w3v[m][r] = ldsP[W3_OFF + m * 16 + 8 * hi + r];
    const float b3 = ldsP[B3_OFF] - 2.19f;

    const v8f vzero = {0.f, 0.f, 0.f, 0.f, 0.f, 0.f, 0.f, 0.f};

    for (int t = 0; t < TPW; ++t) {
        const int tile = blockIdx.x * TILES_WG + t * WAVES + wave;
        const int p    = tile * 16 + pxl;
        const bool pv  = (p < HWPX);

        // ---- stage feats F[66 x 16] (+2 zero pad rows) --------------------
        {
            float c;
            if (hi == 0) c = pv ? (float)(p % WW) * (1.0f / WW) : 0.f;  // w/W
            else         c = pv ? (float)(p / WW) * (1.0f / WW) : 0.f;  // h/W
            stage[2 * pxl + hi] = c;                                    // rows 0,1
            #pragma unroll 8
            for (int row = 2 + hi; row < CIN1; row += 2) {
                float v = pv ? xb[(size_t)(row - 2) * HWPX + p] : 0.f;
                stage[(row >> 1) * 32 + 2 * pxl + (row & 1)] = v;
            }
        }
        __syncthreads();

        // ---- layer 1: H1[64x16] = relu(W1[64x66] * F + b1) ---------------
        v8f acc[4] = {vzero, vzero, vzero, vzero};
        #pragma unroll
        for (int k = 0; k <= 16; ++k) {                     // K = 0..67 (66 real)
            v2f b = *(const v2f*)&stage[(2 * k + hi) * 32 + 2 * pxl];
            #pragma unroll
            for (int m = 0; m < 4; ++m) {
                v2f a = *(const v2f*)&ldsP[W1_OFF + (m * 16 + pxl) * CIN1
                                           + 4 * k + 2 * hi];
                if (k == 16) {                              // zero K=66,67 lanes
                    a.x = hi ? 0.f : a.x;
                    a.y = hi ? 0.f : a.y;
                }
                acc[m] = __builtin_amdgcn_wmma_f32_16x16x4_f32(
                    false, a, false, b, (short)0, acc[m], false, false);
            }
        }
        __syncthreads();

        // bias + relu, restage H1 (overwrites feats; rows 0..63 only)
        #pragma unroll
        for (int m = 0; m < 4; ++m) {
            #pragma unroll
            for (int r = 0; r < 8; ++r) {
                const int ch = m * 16 + 8 * hi + r;
                float v = fmaxf(acc[m][r] + ldsP[B1_OFF + ch], 0.f);
                stage[(ch >> 1) * 32 + 2 * pxl + (ch & 1)] = v;
            }
        }
        __syncthreads();

        // ---- layer 2: H2[64x16] = relu(W2[64x64] * H1 + b2) --------------
        v8f acc2[4] = {vzero, vzero, vzero, vzero};
        #pragma unroll
        for (int k = 0; k < 16; ++k) {
            v2f b = *(const v2f*)&stage[(2 * k + hi) * 32 + 2 * pxl];
            #pragma unroll
            for (int m = 0; m < 4; ++m) {
                v2f a = *(const v2f*)&ldsP[W2_OFF + (m * 16 + pxl) * CH
                                           + 4 * k + 2 * hi];
                acc2[m] = __builtin_amdgcn_wmma_f32_16x16x4_f32(
                    false, a, false, b, (short)0, acc2[m], false, false);
            }
        }

        // ---- layer 3: out[px] = W3 . relu(H2 + b2) + (b3 - 2.19) ---------
        float s = 0.f;
        #pragma unroll
        for (int m = 0; m < 4; ++m)
            #pragma unroll
            for (int r = 0; r < 8; ++r) {
                const int ch = m * 16 + 8 * hi + r;
                float h2 = fmaxf(acc2[m][r] + ldsP[B2_OFF + ch], 0.f);
                s = fmaf(w3v[m][r], h2, s);
            }
        s += __shfl_xor(s, 16, 32);                 // combine the two wave halves
        if (hi == 0 && pv)
            out[(size_t)inst * HWPX + p] = s + b3;
        __syncthreads();                            // WAR guard before next tile
    }
}

extern "C" void kernel_launch(void* const* d_in, const int* in_sizes, int n_in,
                              void* d_out, int out_size, void* d_ws, size_t ws_size,
                              hipStream_t stream) {
    (void)n_in; (void)out_size; (void)d_ws; (void)ws_size;
    const float* x      = (const float*)d_in[0];
    const float* params = (const float*)d_in[1];
    const int*   ni     = (const int*)d_in[2];
    float*       out    = (float*)d_out;

    const int num_insts = in_sizes[1] / NPARAM;   // 32
    const int n_images  = in_sizes[2];            // 8

    dim3 grid(WGS_X, num_insts);
    condlane_mlp_wmma<<<grid, THREADS, 0, stream>>>(x, params, ni, out, n_images);
}